// BHS_GCN_16724602651176
// MI455X (gfx1250) — compile-verified
//
#include <hip/hip_runtime.h>
#include <hip/hip_bf16.h>
#include <stdint.h>

// ---------------------------------------------------------------------------
// BHS_GCN on MI455X (gfx1250, wave32).
//  - fp32 WMMA (v_wmma_f32_16x16x4_f32) for both layer GEMMs, register-blocked
//    8 N-tiles per wave (16x128 output) so the A operand streams once.
//  - fp32 global atomics for the graph scatter (AGG2 fits in 192MB L2).
//  - single fused, LDS-staged, split-K head kernel: weight stream (318MB,
//    read exactly once) is the roofline limiter (~14us @ 23.3TB/s).
// ---------------------------------------------------------------------------

typedef __attribute__((ext_vector_type(2))) float v2f;
typedef __attribute__((ext_vector_type(8))) float v8f;

#define N_NODES 4096
#define BATCH   32

// ---------------- small elementwise / graph-prep kernels -------------------

__global__ void k_zero(float* __restrict__ p, int n) {
    int i = blockIdx.x * blockDim.x + threadIdx.x;
    if (i < n) p[i] = 0.0f;
}

__global__ void k_deg_scatter(const int* __restrict__ dst,
                              const float* __restrict__ ew,
                              float* __restrict__ deg, int E) {
    int e = blockIdx.x * blockDim.x + threadIdx.x;
    if (e < E) atomicAdd(&deg[dst[e]], ew[e]);
}

__global__ void k_deg_finish(const float* __restrict__ deg,
                             float* __restrict__ dinv,
                             float* __restrict__ selfw, int n) {
    int i = blockIdx.x * blockDim.x + threadIdx.x;
    if (i < n) {
        float d = deg[i] + 1.0f;
        dinv[i]  = rsqrtf(d);
        selfw[i] = 1.0f / d;   // dinv^2 : self-loop coefficient
    }
}

__global__ void k_edge_norm(const int* __restrict__ src,
                            const int* __restrict__ dst,
                            const float* __restrict__ ew,
                            const float* __restrict__ dinv,
                            float* __restrict__ norm, int E) {
    int e = blockIdx.x * blockDim.x + threadIdx.x;
    if (e < E) norm[e] = ew[e] * dinv[src[e]] * dinv[dst[e]];
}

// AGG = H * selfw[node]   (scatter accumulator initialized with self-loop term)
__global__ void k_init_self(const float* __restrict__ H,
                            const float* __restrict__ selfw,
                            float* __restrict__ AGG, int C) {
    long long i = (long long)blockIdx.x * blockDim.x + threadIdx.x;
    int n = (int)((i / C) % N_NODES);
    AGG[i] = H[i] * selfw[n];
}

// AGG[b,dst,c] += H[b,src,c] * norm[e]; channel-fastest => coalesced
__global__ void k_edge_scatter(const float* __restrict__ H,
                               float* __restrict__ AGG,
                               const int* __restrict__ src,
                               const int* __restrict__ dst,
                               const float* __restrict__ norm, int C) {
    long long idx = (long long)blockIdx.x * blockDim.x + threadIdx.x;
    int bc  = BATCH * C;
    int e   = (int)(idx / bc);
    int rem = (int)(idx % bc);
    int b = rem / C, c = rem % C;
    int s = src[e], d = dst[e];
    float v = H[((size_t)b * N_NODES + s) * C + c] * norm[e];
    atomicAdd(&AGG[((size_t)b * N_NODES + d) * C + c], v);
}

__global__ void k_bias_relu(const float* __restrict__ AGG,
                            const float* __restrict__ bias,
                            float* __restrict__ out, int C) {
    long long i = (long long)blockIdx.x * blockDim.x + threadIdx.x;
    float v = AGG[i] + bias[(int)(i % C)];
    out[i] = v > 0.0f ? v : 0.0f;
}

// ---------------- fp32 WMMA GEMM: O[M,NOUT] = A[M,KDIM] @ W[KDIM,NOUT] ------
// One wave owns a 16 x (16*NTW) output strip: the A fragment is loaded once
// per k-step and fed to NTW chained v_wmma_f32_16x16x4_f32 ops, cutting the
// A-operand global traffic by NTW (the round-1 layout re-streamed A 8-16x,
// ~1GB for layer 2 = the dominant roofline term). Grids exact => EXEC all-1s.

template <int KDIM, int NOUT, int NTW>
__global__ void k_gemm_wmma(const float* __restrict__ A,
                            const float* __restrict__ W,
                            float* __restrict__ O) {
    constexpr int NGROUP = NOUT / (16 * NTW);   // N-strips per row block
    int tid  = threadIdx.x;
    int wid  = blockIdx.x * (blockDim.x >> 5) + (tid >> 5);
    int mt   = wid / NGROUP;
    int ng   = wid % NGROUP;
    int lane = tid & 31;
    int half = lane >> 4;          // lanes 16-31 hold K+2 / M+8 halves
    int l16  = lane & 15;
    int mrow = mt * 16 + l16;                  // A-operand row for this lane
    int c0   = ng * (16 * NTW) + l16;          // first B-operand col

    v8f acc[NTW];
    #pragma unroll
    for (int t = 0; t < NTW; ++t) acc[t] = (v8f){};

    for (int kk = 0; kk < KDIM; kk += 4) {
        int ka = kk + 2 * half;
        v2f a;
        a.x = A[(size_t)mrow * KDIM + ka];
        a.y = A[(size_t)mrow * KDIM + ka + 1];
        #pragma unroll
        for (int t = 0; t < NTW; ++t) {
            v2f b;
            b.x = W[(size_t)ka       * NOUT + c0 + t * 16];
            b.y = W[(size_t)(ka + 1) * NOUT + c0 + t * 16];
            acc[t] = __builtin_amdgcn_wmma_f32_16x16x4_f32(
                false, a, false, b, (short)0, acc[t], false, false);
        }
    }
    #pragma unroll
    for (int t = 0; t < NTW; ++t)
        #pragma unroll
        for (int r = 0; r < 8; ++r)
            O[(size_t)(mt * 16 + r + 8 * half) * NOUT + c0 + t * 16] = acc[t][r];
}

// ---------------- fused head GEMM: [32,K] @ {v1W[K,64] | advW[K,12]} --------
// Split-K. The feat tile [32 x KT] is staged in LDS once per block (feat
// global traffic = 134MB exactly); columns 0..63 -> v1W, 64..75 -> advW.
// Weight loads coalesced and read exactly once: 318MB = roofline limiter.

#define HEAD_KT 128

__global__ void k_head_fused(const float* __restrict__ feat,
                             const float* __restrict__ v1W,
                             const float* __restrict__ advW,
                             float* __restrict__ accv1,
                             float* __restrict__ accadv, int K, int KC) {
    __shared__ float sf[BATCH * HEAD_KT];
    int t  = threadIdx.x;
    int j  = t & 127;          // column lane (0..127; 0..63 v1, 64..75 adv)
    int kl = t >> 7;           // 2 k-lanes
    int kbase = blockIdx.x * KC;

    float acc[BATCH];
    #pragma unroll
    for (int b = 0; b < BATCH; ++b) acc[b] = 0.0f;

    for (int k0 = kbase; k0 < kbase + KC; k0 += HEAD_KT) {
        __syncthreads();
        for (int i = t; i < BATCH * HEAD_KT; i += 256) {
            int b = i / HEAD_KT, kk = i % HEAD_KT;
            sf[i] = feat[(size_t)b * K + k0 + kk];
        }
        __syncthreads();
        for (int kk = kl; kk < HEAD_KT; kk += 2) {
            float w = 0.0f;
            if (j < 64)       w = v1W[(size_t)(k0 + kk) * 64 + j];
            else if (j < 76)  w = advW[(size_t)(k0 + kk) * 12 + (j - 64)];
            #pragma unroll
            for (int b = 0; b < BATCH; ++b)
                acc[b] += sf[b * HEAD_KT + kk] * w;
        }
    }
    if (j < 64) {
        #pragma unroll
        for (int b = 0; b < BATCH; ++b) atomicAdd(&accv1[b * 64 + j], acc[b]);
    } else if (j < 76) {
        #pragma unroll
        for (int b = 0; b < BATCH; ++b) atomicAdd(&accadv[b * 12 + (j - 64)], acc[b]);
    }
}

// ---------------- tiny dense tail: value MLP + dueling combine --------------

__global__ void k_head_final(const float* __restrict__ accv1,
                             const float* __restrict__ accadv,
                             const float* __restrict__ v1b,
                             const float* __restrict__ v2W,
                             const float* __restrict__ v2b,
                             const float* __restrict__ v3W,
                             const float* __restrict__ v3b,
                             const float* __restrict__ advb,
                             float* __restrict__ out) {
    __shared__ float s_v1[BATCH * 64];
    __shared__ float s_v2[BATCH * 64];
    __shared__ float s_adv[BATCH * 12];
    __shared__ float s_v3[BATCH];
    int t = threadIdx.x;

    for (int i = t; i < BATCH * 64; i += 256) {
        float v = accv1[i] + v1b[i & 63];
        s_v1[i] = v > 0.0f ? v : 0.0f;
    }
    for (int i = t; i < BATCH * 12; i += 256) {
        float v = accadv[i] + advb[i % 12];
        s_adv[i] = v > 0.0f ? v : 0.0f;
    }
    __syncthreads();
    for (int i = t; i < BATCH * 64; i += 256) {
        int b = i >> 6, j = i & 63;
        float s = v2b[j];
        for (int k = 0; k < 64; ++k) s += s_v1[b * 64 + k] * v2W[k * 64 + j];
        s_v2[i] = s > 0.0f ? s : 0.0f;
    }
    __syncthreads();
    if (t < BATCH) {
        float s = v3b[0];
        for (int k = 0; k < 64; ++k) s += s_v2[t * 64 + k] * v3W[k];
        s_v3[t] = s;
    }
    __syncthreads();
    for (int i = t; i < BATCH * 12; i += 256) {
        int b = i / 12, r = i % 12, h = r >> 2;
        const float* a = &s_adv[b * 12 + h * 4];
        float m = 0.25f * (a[0] + a[1] + a[2] + a[3]);
        out[i] = s_v3[b] + s_adv[i] - m;
    }
}

// ---------------------------------------------------------------------------

extern "C" void kernel_launch(void* const* d_in, const int* in_sizes, int n_in,
                              void* d_out, int out_size, void* d_ws, size_t ws_size,
                              hipStream_t stream) {
    (void)n_in; (void)out_size; (void)ws_size;
    const float* x    = (const float*)d_in[0];               // [32,4096,16]
    const int*   eidx = (const int*)d_in[1];                 // [2,E] int
    const float* ew   = (const float*)d_in[2];               // [E]
    const float* W1   = (const float*)d_in[3];               // [16,128]
    const float* b1   = (const float*)d_in[4];
    const float* W2   = (const float*)d_in[5];               // [128,256]
    const float* b2   = (const float*)d_in[6];
    const float* advW = (const float*)d_in[7];               // [1048576,12]
    const float* advb = (const float*)d_in[8];
    const float* v1W  = (const float*)d_in[9];               // [1048576,64]
    const float* v1b  = (const float*)d_in[10];
    const float* v2W  = (const float*)d_in[11];              // [64,64]
    const float* v2b  = (const float*)d_in[12];
    const float* v3W  = (const float*)d_in[13];              // [64,1]
    const float* v3b  = (const float*)d_in[14];
    float* out = (float*)d_out;                              // [32,3,4]

    const int E = in_sizes[1] / 2;
    const int* src = eidx;
    const int* dst = eidx + E;

    const size_t M   = (size_t)BATCH * N_NODES;              // 131072 rows
    const size_t NF1 = M * 128;                              // 16,777,216 floats
    const size_t NF2 = M * 256;                              // 33,554,432 floats

    // Workspace layout (floats). AGG2 aliases [H1|AGG1] which are dead by then.
    float* ws     = (float*)d_ws;
    float* H1     = ws;                  // 67 MB (H1R after bias+relu)
    float* AGG1   = ws + NF1;            // 67 MB
    float* AGG2   = ws;                  // 134 MB alias of H1+AGG1
    float* H2     = ws + 2 * NF1;        // 134 MB (H2pre, then feat)
    float* deg    = H2 + NF2;
    float* dinv   = deg  + N_NODES;
    float* selfw  = dinv + N_NODES;
    float* norm   = selfw + N_NODES;     // [E]
    float* accv1  = norm + E;            // [32*64]
    float* accadv = accv1 + BATCH * 64;  // [32*12]

    // ---- graph normalization -------------------------------------------
    k_zero<<<(N_NODES + 255) / 256, 256, 0, stream>>>(deg, N_NODES);
    k_zero<<<(BATCH * 64 + 255) / 256, 256, 0, stream>>>(accv1, BATCH * 64);
    k_zero<<<(BATCH * 12 + 255) / 256, 256, 0, stream>>>(accadv, BATCH * 12);
    k_deg_scatter<<<(E + 255) / 256, 256, 0, stream>>>(dst, ew, deg, E);
    k_deg_finish<<<(N_NODES + 255) / 256, 256, 0, stream>>>(deg, dinv, selfw, N_NODES);
    k_edge_norm<<<(E + 255) / 256, 256, 0, stream>>>(src, dst, ew, dinv, norm, E);

    // ---- layer 1: GEMM (WMMA, 16x128 strip/wave) + scatter + bias/ReLU --
    {
        int waves = (int)((M / 16) * (128 / (16 * 8)));      // 8192
        k_gemm_wmma<16, 128, 8><<<waves / 8, 256, 0, stream>>>(x, W1, H1);
    }
    k_init_self<<<(int)(NF1 / 256), 256, 0, stream>>>(H1, selfw, AGG1, 128);
    {
        long long tot = (long long)E * BATCH * 128;
        k_edge_scatter<<<(int)(tot / 256), 256, 0, stream>>>(H1, AGG1, src, dst, norm, 128);
    }
    k_bias_relu<<<(int)(NF1 / 256), 256, 0, stream>>>(AGG1, b1, H1, 128); // H1 := H1R

    // ---- layer 2: GEMM (WMMA, 16x128 strip/wave) + scatter + bias/ReLU --
    {
        int waves = (int)((M / 16) * (256 / (16 * 8)));      // 16384
        k_gemm_wmma<128, 256, 8><<<waves / 8, 256, 0, stream>>>(H1, W2, H2);
    }
    k_init_self<<<(int)(NF2 / 256), 256, 0, stream>>>(H2, selfw, AGG2, 256);
    {
        long long tot = (long long)E * BATCH * 256;
        k_edge_scatter<<<(int)(tot / 256), 256, 0, stream>>>(H2, AGG2, src, dst, norm, 256);
    }
    k_bias_relu<<<(int)(NF2 / 256), 256, 0, stream>>>(AGG2, b2, H2, 256); // H2 := feat

    // ---- fused, LDS-staged, memory-bound head GEMMs ---------------------
    const int K  = N_NODES * 256;    // 1,048,576
    const int KC = 4096;             // 256 split-K slices
    k_head_fused<<<K / KC, 256, 0, stream>>>(H2, v1W, advW, accv1, accadv, K, KC);

    // ---- tiny dense tail -------------------------------------------------
    k_head_final<<<1, 256, 0, stream>>>(accv1, accadv, v1b, v2W, v2b, v3W, v3b,
                                        advb, out);
}